// LayerOut2LayerIn_43404939493623
// MI455X (gfx1250) — compile-verified
//
#include <hip/hip_runtime.h>

// LayerOut2LayerIn: new[i] = (ds_in[i] + segment_sum(ds_out[src], recv)) / (1 + indeg)
// MI455X plan: whole working set (ds_out 25.6MB, accumulator 25.6MB) lives in 192MB L2;
// bottleneck is L2 atomic-f32 throughput. Thread-per-edge scatter with b128 gathers,
// no-return f32 atomics, async global->LDS staging of the edge-index stream.
// (k_scatter is defined first so the disasm snippet shows the hot kernel.)

#define BLK 256

typedef float v4f __attribute__((ext_vector_type(4)));

#define AS_GLOBAL __attribute__((address_space(1)))
#define AS_LDS    __attribute__((address_space(3)))

#if defined(__HIP_DEVICE_COMPILE__) && \
    __has_builtin(__builtin_amdgcn_global_load_async_to_lds_b32) && \
    __has_builtin(__builtin_amdgcn_s_wait_asynccnt)
#define USE_ASYNC_LDS 1
#else
#define USE_ASYNC_LDS 0
#endif

// ---------------------------------------------------------------------------
// Edge scatter: msg[recv[e]] += ds_out[src[e]]; deg[recv[e]] += 1.
// Thread-per-edge, grid-stride tiles of BLK edges. Edge indices are staged
// through LDS with gfx1250 async copies (double buffered, per-lane
// self-contained: each lane consumes exactly the LDS slot its own async
// copy wrote, so no workgroup barrier is needed).
// ---------------------------------------------------------------------------
__global__ __launch_bounds__(BLK) void k_scatter(
    const int*   __restrict__ src,
    const int*   __restrict__ recv,
    const float* __restrict__ ds_out,
    float*       __restrict__ msg,
    float*       __restrict__ deg,
    int nE)
{
    __shared__ int s_idx[2][2][BLK];   // [buffer][src|recv][lane]

    const int tid        = threadIdx.x;
    const int tileStride = gridDim.x * BLK;

    auto stage = [&](int base, int buf) {
        int e = base + tid;
        if (e >= nE) e = nE - 1;       // clamp: keep EXEC full so ASYNCcnt accounting is exact
#if USE_ASYNC_LDS
        __builtin_amdgcn_global_load_async_to_lds_b32(
            (AS_GLOBAL int*)(src + e),  (AS_LDS int*)&s_idx[buf][0][tid], 0, 0);
        __builtin_amdgcn_global_load_async_to_lds_b32(
            (AS_GLOBAL int*)(recv + e), (AS_LDS int*)&s_idx[buf][1][tid], 0, 0);
#else
        s_idx[buf][0][tid] = src[e];
        s_idx[buf][1][tid] = recv[e];
#endif
    };

    int base = blockIdx.x * BLK;
    if (base >= nE) return;            // uniform per block

    stage(base, 0);
    int buf = 0;

    for (; base < nE; base += tileStride) {
        int nxt = base + tileStride;   // uniform
        if (nxt < nE) {
            stage(nxt, buf ^ 1);       // prefetch next tile's indices
#if USE_ASYNC_LDS
            __builtin_amdgcn_s_wait_asynccnt(2);   // current tile's 2 copies retired (in-order)
#endif
        } else {
#if USE_ASYNC_LDS
            __builtin_amdgcn_s_wait_asynccnt(0);
#endif
        }

        int e = base + tid;
        if (e < nE) {
            const int s = s_idx[buf][0][tid];
            const int r = s_idx[buf][1][tid];

            const v4f* __restrict__ srow = (const v4f*)ds_out + (size_t)s * 16;
            float*     __restrict__ mrow = msg + (size_t)r * 64;

            #pragma unroll
            for (int c = 0; c < 16; ++c) {
                v4f v = srow[c];                   // global_load_b128 gather (L2-resident)
                atomicAdd(mrow + 4 * c + 0, v.x);  // no-return f32 atomics -> STOREcnt
                atomicAdd(mrow + 4 * c + 1, v.y);
                atomicAdd(mrow + 4 * c + 2, v.z);
                atomicAdd(mrow + 4 * c + 3, v.w);
            }
            atomicAdd(deg + r, 1.0f);
        }
        buf ^= 1;
    }
}

// ---------------------------------------------------------------------------
// Zero the workspace accumulator (msg[N*64] ++ deg[N], contiguous floats).
// ---------------------------------------------------------------------------
__global__ __launch_bounds__(BLK) void k_zero(float* __restrict__ p, long long n)
{
    long long i  = (long long)blockIdx.x * BLK + threadIdx.x;
    long long n4 = n >> 2;
    if (i < n4) {
        v4f z = {0.0f, 0.0f, 0.0f, 0.0f};
        __builtin_nontemporal_store(z, (v4f*)p + i);
    }
    if (i < (n & 3)) {                 // scalar tail
        p[(n4 << 2) + i] = 0.0f;
    }
}

// ---------------------------------------------------------------------------
// Finalize: out = (ds_in + msg) * 1/(deg + 1).  One float4 per thread.
// Single-use streams -> non-temporal; deg is shared by 16 adjacent threads.
// ---------------------------------------------------------------------------
__global__ __launch_bounds__(BLK) void k_finalize(
    const float* __restrict__ ds_in,
    const float* __restrict__ msg,
    const float* __restrict__ deg,
    float*       __restrict__ out,
    int nNodes)
{
    int t     = blockIdx.x * BLK + threadIdx.x;
    int total = nNodes * 16;           // float4 chunks
    if (t >= total) return;

    int   node = t >> 4;
    float d    = deg[node];
    float inv  = 1.0f / (d + 1.0f);

    v4f a = __builtin_nontemporal_load((const v4f*)ds_in + t);
    v4f m = __builtin_nontemporal_load((const v4f*)msg  + t);
    v4f r = (a + m) * inv;
    __builtin_nontemporal_store(r, (v4f*)out + t);
}

// ---------------------------------------------------------------------------
extern "C" void kernel_launch(void* const* d_in, const int* in_sizes, int n_in,
                              void* d_out, int out_size, void* d_ws, size_t ws_size,
                              hipStream_t stream)
{
    const float* ds_in  = (const float*)d_in[0];
    const float* ds_out = (const float*)d_in[1];
    const int*   edges  = (const int*)d_in[2];     // [2, E] row-major: [src | recv]

    const int nNodes = in_sizes[0] / 64;
    const int nE     = in_sizes[2] / 2;

    const int* src  = edges;
    const int* recv = edges + nE;

    float* msg = (float*)d_ws;                         // [nNodes * 64]
    float* deg = msg + (size_t)nNodes * 64;            // [nNodes]
    float* out = (float*)d_out;

    // 1) zero accumulator (ws is poisoned by harness; must re-zero every call)
    long long zn = (long long)nNodes * 65;
    int zb = (int)((zn / 4 + BLK - 1) / BLK) + 1;
    k_zero<<<zb, BLK, 0, stream>>>(msg, zn);

    // 2) scatter-add over edges (grid-stride so each block pipelines >1 tile)
    int tiles = (nE + BLK - 1) / BLK;
    int sb    = tiles < 1568 ? tiles : 1568;
    k_scatter<<<sb, BLK, 0, stream>>>(src, recv, ds_out, msg, deg, nE);

    // 3) finalize
    int fb = (nNodes * 16 + BLK - 1) / BLK;
    k_finalize<<<fb, BLK, 0, stream>>>(ds_in, msg, deg, out, nNodes);
}